// GGAELoss_42563125903675
// MI455X (gfx1250) — compile-verified
//
#include <hip/hip_runtime.h>
#include <hip/hip_bf16.h>

// Problem constants (from reference)
#define B_N   4096          // batch size B
#define NF    16384         // K_full dimension
#define ZD    8             // latent dim n
#define COLS  80            // 64 (outer) + 8 (z) + 8 (pad) GEMM columns
#define INV_CB 0.08f        // 1 / (c * bandwidth) = 1 / (0.25 * 50)
#define EPSF  1e-8f

typedef float v2f __attribute__((ext_vector_type(2)));
typedef float v8f __attribute__((ext_vector_type(8)));

// ---------------------------------------------------------------------------
// Kernel 1: double-gather K_batch[i][j] = K_full[idx[i], idx[j]] and row sums
//           invd[i] = 1 / (sum_j K_batch[i][j] + eps)
// ---------------------------------------------------------------------------
__global__ void __launch_bounds__(256)
k_gather_d(const float* __restrict__ Kf, const int* __restrict__ idx,
           float* __restrict__ Kb, float* __restrict__ invd) {
  __shared__ float red[256];
  const int i = blockIdx.x;
  const size_t rowbase = (size_t)idx[i] * NF;
  float s = 0.f;
  for (int j = threadIdx.x; j < B_N; j += 256) {
    float v = Kf[rowbase + (size_t)idx[j]];
    Kb[(size_t)i * B_N + j] = v;
    s += v;
  }
  red[threadIdx.x] = s;
  __syncthreads();
  for (int off = 128; off > 0; off >>= 1) {
    if (threadIdx.x < off) red[threadIdx.x] += red[threadIdx.x + off];
    __syncthreads();
  }
  if (threadIdx.x == 0) invd[i] = 1.0f / (red[0] + EPSF);
}

// ---------------------------------------------------------------------------
// Kernel 2: invdt[i] = 1 / (invd[i] * sum_j Kb[i][j]*invd[j] + eps)
// ---------------------------------------------------------------------------
__global__ void __launch_bounds__(256)
k_dtilde(const float* __restrict__ Kb, const float* __restrict__ invd,
         float* __restrict__ invdt) {
  __shared__ float red[256];
  const int i = blockIdx.x;
  const float* __restrict__ row = Kb + (size_t)i * B_N;
  float s = 0.f;
  for (int j = threadIdx.x; j < B_N; j += 256) s += row[j] * invd[j];
  red[threadIdx.x] = s;
  __syncthreads();
  for (int off = 128; off > 0; off >>= 1) {
    if (threadIdx.x < off) red[threadIdx.x] += red[threadIdx.x + off];
    __syncthreads();
  }
  if (threadIdx.x == 0) invdt[i] = 1.0f / (red[0] * invd[i] + EPSF);
}

// ---------------------------------------------------------------------------
// Kernel 3: build GEMM right-hand panel Bm (B_N x COLS), row-major:
//   cols 0..63 : z[j][c/8] * z[j][c%8]   (flattened outer product)
//   cols 64..71: z[j][c-64]
//   cols 72..79: 0 (pad)
// ---------------------------------------------------------------------------
__global__ void __launch_bounds__(256)
k_buildB(const float* __restrict__ z, float* __restrict__ Bm) {
  int t = blockIdx.x * 256 + threadIdx.x;
  if (t >= B_N * COLS) return;
  int j = t / COLS, c = t % COLS;
  float v = 0.f;
  if (c < 64)       v = z[j * ZD + (c >> 3)] * z[j * ZD + (c & 7)];
  else if (c < 72)  v = z[j * ZD + (c - 64)];
  Bm[t] = v;
}

// ---------------------------------------------------------------------------
// Kernel 4: WMMA GEMM  Out = L @ Bm   (4096x4096 @ 4096x80, f32)
//   L_ij = Kb_ij*invd_i*invd_j*invdt_i*INV_CB - (i==j)*INV_CB  (fused on load)
// One wave (32 lanes) per block; block b owns M-tile rows [16b, 16b+16).
// Uses V_WMMA_F32_16X16X4_F32: A=v2f (16x4), B=v2f (4x16), C/D=v8f (16x16).
// ---------------------------------------------------------------------------
__global__ void __launch_bounds__(32)
k_gemm(const float* __restrict__ Kb, const float* __restrict__ invd,
       const float* __restrict__ invdt, const float* __restrict__ Bm,
       float* __restrict__ Out) {
  const int lane = threadIdx.x;           // 0..31
  const int m0   = blockIdx.x * 16;
  const int M    = lane & 15;             // row within tile (A) / col N (B,C)
  const int koff = (lane >> 4) * 2;       // K sub-offset for this half-wave
  const int row  = m0 + M;

  const float rs = invd[row] * invdt[row] * INV_CB;   // row scale of L
  const float* __restrict__ Arow = Kb + (size_t)row * B_N;

  v8f c0 = {}, c1 = {}, c2 = {}, c3 = {}, c4 = {};

  for (int kk = 0; kk < B_N; kk += 4) {
    const int j0 = kk + koff;
    // A fragment: L[row][j0], L[row][j0+1] built on the fly
    v2f a;
    a.x = Arow[j0]     * rs * invd[j0]     - (row == j0     ? INV_CB : 0.f);
    a.y = Arow[j0 + 1] * rs * invd[j0 + 1] - (row == j0 + 1 ? INV_CB : 0.f);
    // B fragments for the 5 N-tiles: B[j0..j0+1][t*16 + M]
    const float* __restrict__ Bp = Bm + (size_t)j0 * COLS + M;
    v2f b0, b1, b2, b3, b4;
    b0.x = Bp[0];  b0.y = Bp[COLS];
    b1.x = Bp[16]; b1.y = Bp[COLS + 16];
    b2.x = Bp[32]; b2.y = Bp[COLS + 32];
    b3.x = Bp[48]; b3.y = Bp[COLS + 48];
    b4.x = Bp[64]; b4.y = Bp[COLS + 64];

    c0 = __builtin_amdgcn_wmma_f32_16x16x4_f32(false, a, false, b0, (short)0, c0, false, false);
    c1 = __builtin_amdgcn_wmma_f32_16x16x4_f32(false, a, false, b1, (short)0, c1, false, false);
    c2 = __builtin_amdgcn_wmma_f32_16x16x4_f32(false, a, false, b2, (short)0, c2, false, false);
    c3 = __builtin_amdgcn_wmma_f32_16x16x4_f32(false, a, false, b3, (short)0, c3, false, false);
    c4 = __builtin_amdgcn_wmma_f32_16x16x4_f32(false, a, false, b4, (short)0, c4, false, false);
  }

  // Store C/D: VGPR v holds row (v + 8*(lane>>4)), col = t*16 + (lane&15)
  const int rbase = m0 + ((lane >> 4) * 8);
  const int nbase = lane & 15;
#pragma unroll
  for (int v = 0; v < 8; ++v) {
    float* __restrict__ o = Out + (size_t)(rbase + v) * COLS + nbase;
    o[0]  = c0[v];
    o[16] = c1[v];
    o[32] = c2[v];
    o[48] = c3[v];
    o[64] = c4[v];
  }
}

// ---------------------------------------------------------------------------
// Kernel 5: per-row loss, block-level partial sums (deterministic)
//   Out row i: T1 = cols 0..63 (8x8), LY = cols 64..71
//   H_ab = 0.5*(T1_ab - z_a*LY_b - LY_a*z_b)
//   loss_i = sum_ab H_ab*H_ba - 2*trace(H) + n
// ---------------------------------------------------------------------------
__global__ void __launch_bounds__(256)
k_rowloss(const float* __restrict__ Out, const float* __restrict__ z,
          float* __restrict__ partial) {
  __shared__ float red[256];
  const int i = blockIdx.x * 256 + threadIdx.x;   // grid covers exactly B_N
  const float* __restrict__ T = Out + (size_t)i * COLS;
  float zi[ZD], LY[ZD], H[ZD][ZD];
#pragma unroll
  for (int a = 0; a < ZD; ++a) { zi[a] = z[i * ZD + a]; LY[a] = T[64 + a]; }
#pragma unroll
  for (int a = 0; a < ZD; ++a)
#pragma unroll
    for (int b = 0; b < ZD; ++b)
      H[a][b] = 0.5f * (T[a * ZD + b] - zi[a] * LY[b] - LY[a] * zi[b]);
  float trH = 0.f, trH2 = 0.f;
#pragma unroll
  for (int a = 0; a < ZD; ++a) trH += H[a][a];
#pragma unroll
  for (int a = 0; a < ZD; ++a)
#pragma unroll
    for (int b = 0; b < ZD; ++b) trH2 += H[a][b] * H[b][a];
  red[threadIdx.x] = trH2 - 2.f * trH + (float)ZD;
  __syncthreads();
  for (int off = 128; off > 0; off >>= 1) {
    if (threadIdx.x < off) red[threadIdx.x] += red[threadIdx.x + off];
    __syncthreads();
  }
  if (threadIdx.x == 0) partial[blockIdx.x] = red[0];
}

// Kernel 6: reduce the 16 block partials -> mean
__global__ void k_final(const float* __restrict__ partial, float* __restrict__ out) {
  if (threadIdx.x == 0) {
    float s = 0.f;
    for (int b = 0; b < B_N / 256; ++b) s += partial[b];
    out[0] = s * (1.0f / (float)B_N);
  }
}

// ---------------------------------------------------------------------------
extern "C" void kernel_launch(void* const* d_in, const int* in_sizes, int n_in,
                              void* d_out, int out_size, void* d_ws, size_t ws_size,
                              hipStream_t stream) {
  const float* z   = (const float*)d_in[0];   // (4096, 8)
  const float* Kf  = (const float*)d_in[1];   // (16384, 16384)
  const int*   idx = (const int*)d_in[2];     // (4096,)
  float* out = (float*)d_out;

  char* ws = (char*)d_ws;
  size_t off = 0;
  float* Kb     = (float*)(ws + off); off += (size_t)B_N * B_N * sizeof(float);
  float* invd   = (float*)(ws + off); off += (size_t)B_N * sizeof(float);
  float* invdt  = (float*)(ws + off); off += (size_t)B_N * sizeof(float);
  float* Bm     = (float*)(ws + off); off += (size_t)B_N * COLS * sizeof(float);
  float* Outp   = (float*)(ws + off); off += (size_t)B_N * COLS * sizeof(float);
  float* part   = (float*)(ws + off); off += (B_N / 256) * sizeof(float);
  (void)ws_size; (void)in_sizes; (void)n_in; (void)out_size;

  k_gather_d<<<B_N, 256, 0, stream>>>(Kf, idx, Kb, invd);
  k_dtilde  <<<B_N, 256, 0, stream>>>(Kb, invd, invdt);
  k_buildB  <<<(B_N * COLS + 255) / 256, 256, 0, stream>>>(z, Bm);
  k_gemm    <<<B_N / 16, 32, 0, stream>>>(Kb, invd, invdt, Bm, Outp);
  k_rowloss <<<B_N / 256, 256, 0, stream>>>(Outp, z, part);
  k_final   <<<1, 32, 0, stream>>>(part, out);
}